// BestMatchDistance_47485158425242
// MI455X (gfx1250) — compile-verified
//
#include <hip/hip_runtime.h>

// ---------------------------------------------------------------------------
// BestMatchDistance on MI455X (gfx1250, wave32)
//   sim[b,q,s] = sum_d Q[b,d,q]*S[b,d,s];  out[b] = mean_q max_s sim[b,q,s]
// Batched GEMM via V_WMMA_F32_16X16X4_F32 (exact fp32 matrix path).
// S staged through LDS *column-major* (transposed during the async copy) so
// the B-operand K-pair is one aligned ds_load_b64 straight into a VGPR pair.
// ---------------------------------------------------------------------------

#define B_N   64
#define D_N   64
#define NQ_N  2048
#define NS_N  2048

#define WAVES   8                  // 256 threads / 32
#define QPW     32                 // q rows per wave (2 x 16-row A tiles)
#define QPB     (WAVES * QPW)      // 256 q rows per block
#define QTILES  (NQ_N / QPB)       // 8 blocks per batch
#define SW      64                 // staged S columns per stage
#define NSTAGE  (NS_N / SW)        // 32 stages
#define CSTR    68                 // LDS floats per staged column (64 d + 4 pad)
                                   // read bank = (4*col + d) % 64 -> conflict-free

typedef __attribute__((ext_vector_type(2))) float v2f;
typedef __attribute__((ext_vector_type(8))) float v8f;

typedef __attribute__((address_space(1))) int* gi32_ptr;   // global i32
typedef __attribute__((address_space(3))) int* li32_ptr;   // LDS i32

#if defined(__has_builtin)
#if __has_builtin(__builtin_amdgcn_global_load_async_to_lds_b32)
#define USE_ASYNC_LDS 1
#endif
#endif

__global__ __launch_bounds__(256)
void bmd_main_kernel(const float* __restrict__ Q, const float* __restrict__ S,
                     float* __restrict__ part) {
  __shared__ float sm[2][SW * CSTR];    // 2 x 17 KB staged S tiles (col-major)
  __shared__ float wsum[WAVES];

  const int b    = blockIdx.x >> 3;     // NQ_N/QPB == 8 q-tiles per batch
  const int qt   = blockIdx.x & 7;
  const int q0   = qt * QPB;
  const int tid  = threadIdx.x;
  const int w    = tid >> 5;
  const int lane = tid & 31;
  const int half = lane >> 4;           // which 16-lane half
  const int l    = lane & 15;

  const float* Qb = Q + (size_t)b * D_N * NQ_N;
  const float* Sb = S + (size_t)b * D_N * NS_N;

  // ---- Load two 16x64 A tiles per wave into registers (ISA fp32 A layout:
  //      lane M=l, VGPR pair holds K = 2*half, 2*half+1 within each K-quad).
  v2f a0[16], a1[16];
  const int qr = q0 + w * QPW + l;
#pragma unroll
  for (int kk = 0; kk < 16; ++kk) {
    const int d0 = 4 * kk + 2 * half;
    a0[kk].x = Qb[(size_t)d0 * NQ_N + qr];
    a0[kk].y = Qb[(size_t)(d0 + 1) * NQ_N + qr];
    a1[kk].x = Qb[(size_t)d0 * NQ_N + qr + 16];
    a1[kk].y = Qb[(size_t)(d0 + 1) * NQ_N + qr + 16];
  }

  float rm0[8], rm1[8];
#pragma unroll
  for (int r = 0; r < 8; ++r) { rm0[r] = -__builtin_inff(); rm1[r] = -__builtin_inff(); }

  // ---- Stage loader: S[b, 0:64, c0:c0+64] -> sm[buf] transposed (col-major).
  // Per wave-instruction chunk: 2 d-rows x 16 cols -> global: two 64B
  // segments; LDS write bank = 4*col + d -> conflict-free.
  auto load_stage = [&](int stage, int buf) {
    const int c0 = stage * SW;
#pragma unroll
    for (int j = 0; j < 16; ++j) {
      const int ws  = j * 8 + w;                 // waveslot 0..127
      const int d   = ((ws >> 2) << 1) + half;   // 2-row pair + lane half
      const int col = ((ws & 3) << 4) + l;       // 16-col group + lane
      const float* gp = Sb + (size_t)d * NS_N + c0 + col;
      float* lp = &sm[buf][col * CSTR + d];
#ifdef USE_ASYNC_LDS
      __builtin_amdgcn_global_load_async_to_lds_b32(
          (gi32_ptr)gp, (li32_ptr)lp, 0, 0);
#else
      *lp = *gp;
#endif
    }
  };

  load_stage(0, 0);

  for (int st = 0; st < NSTAGE; ++st) {
    const int buf = st & 1;
#ifdef USE_ASYNC_LDS
    asm volatile("s_wait_asynccnt 0x0" ::: "memory");
#endif
    __syncthreads();                          // sm[buf] ready for all waves
    if (st + 1 < NSTAGE) load_stage(st + 1, buf ^ 1);   // prefetch next stage

#pragma unroll
    for (int sst = 0; sst < 4; ++sst) {       // 16-column output subtiles
      // One base per subtile; kk reads become small immediate DS offsets.
      const float* colbase = &sm[buf][(sst * 16 + l) * CSTR];
      v8f acc0 = {0.f, 0.f, 0.f, 0.f, 0.f, 0.f, 0.f, 0.f};
      v8f acc1 = {0.f, 0.f, 0.f, 0.f, 0.f, 0.f, 0.f, 0.f};
#pragma unroll
      for (int kk = 0; kk < 16; ++kk) {       // K = 64 in steps of 4
        const int d0 = 4 * kk + 2 * half;
        const v2f bv = *(const v2f*)(colbase + d0);   // one aligned b64
        acc0 = __builtin_amdgcn_wmma_f32_16x16x4_f32(
            false, a0[kk], false, bv, (short)0, acc0, false, false);
        acc1 = __builtin_amdgcn_wmma_f32_16x16x4_f32(
            false, a1[kk], false, bv, (short)0, acc1, false, false);
      }
#pragma unroll
      for (int r = 0; r < 8; ++r) {
        rm0[r] = fmaxf(rm0[r], acc0[r]);
        rm1[r] = fmaxf(rm1[r], acc1[r]);
      }
    }
    __syncthreads();                          // everyone done reading sm[buf]
  }

  // ---- Max across the 16 lanes sharing each output row (wave32 shuffles).
#pragma unroll
  for (int m = 1; m <= 8; m <<= 1) {
#pragma unroll
    for (int r = 0; r < 8; ++r) {
      rm0[r] = fmaxf(rm0[r], __shfl_xor(rm0[r], m, 32));
      rm1[r] = fmaxf(rm1[r], __shfl_xor(rm1[r], m, 32));
    }
  }
  // halves hold disjoint row sets -> sum both halves' contributions
  float ls = 0.f;
#pragma unroll
  for (int r = 0; r < 8; ++r) ls += rm0[r] + rm1[r];
  ls += __shfl_xor(ls, 16, 32);               // 32 rows of this wave total

  if (lane == 0) wsum[w] = ls;
  __syncthreads();
  if (tid == 0) {
    float p = 0.f;
#pragma unroll
    for (int i = 0; i < WAVES; ++i) p += wsum[i];
    part[blockIdx.x] = p;                     // sum of 256 row-maxes
  }
}

// Deterministic final mean: no float atomics anywhere.
__global__ void bmd_final_kernel(const float* __restrict__ part,
                                 float* __restrict__ out) {
  const int t = threadIdx.x;
  if (t < B_N) {
    float s = 0.f;
#pragma unroll
    for (int j = 0; j < QTILES; ++j) s += part[t * QTILES + j];
    out[t] = s * (1.0f / NQ_N);
  }
}

extern "C" void kernel_launch(void* const* d_in, const int* in_sizes, int n_in,
                              void* d_out, int out_size, void* d_ws, size_t ws_size,
                              hipStream_t stream) {
  (void)in_sizes; (void)n_in; (void)out_size; (void)ws_size;
  const float* Q = (const float*)d_in[0];   // (B, D, NQ) fp32
  const float* S = (const float*)d_in[1];   // (B, D, NS) fp32
  float* out  = (float*)d_out;              // (B,) fp32
  float* part = (float*)d_ws;               // B*QTILES = 512 floats of scratch

  bmd_main_kernel<<<B_N * QTILES, 256, 0, stream>>>(Q, S, part);
  bmd_final_kernel<<<1, 64, 0, stream>>>(part, out);
}